// GraphEmbeddings_66073776881702
// MI455X (gfx1250) — compile-verified
//
#include <hip/hip_runtime.h>
#include <stdint.h>

#define NP       32768
#define MNBR     12
#define DDIM     64
#define HID      768         // MNBR * DDIM
#define BCRY     64
#define PATM     512
#define CAP_O    180
#define CAP_C    120
#define SLOTS    300
#define ROWS     (BCRY * SLOTS)       // 19200
#define EMB_ROWS 119
#define EMB_F    (EMB_ROWS * DDIM)    // 7616 floats per table
#define SMEM_F   (2 * EMB_F)          // 15232 floats = 60928 B LDS
#define F4_TAB   (EMB_F / 4)          // 1904 float4 per table
#define F4_TOT   (2 * F4_TAB)         // 3808

// ---------------------------------------------------------------------------
// Threefry-2x32-20 (JAX PRNG family). Rotations {13,15,26,6} / {17,29,16,24},
// key injection every 4 rounds.
// ---------------------------------------------------------------------------
__device__ __forceinline__ uint32_t rotl32(uint32_t x, int r) {
  return (x << r) | (x >> (32 - r));
}

__device__ __forceinline__ void threefry2x32(uint32_t k0, uint32_t k1,
                                             uint32_t c0, uint32_t c1,
                                             uint32_t& o0, uint32_t& o1) {
  uint32_t ks0 = k0, ks1 = k1, ks2 = 0x1BD11BDAu ^ k0 ^ k1;
  uint32_t x0 = c0 + ks0, x1 = c1 + ks1;
#define TF_R(r) { x0 += x1; x1 = rotl32(x1, (r)); x1 ^= x0; }
  TF_R(13) TF_R(15) TF_R(26) TF_R(6)   x0 += ks1; x1 += ks2 + 1u;
  TF_R(17) TF_R(29) TF_R(16) TF_R(24)  x0 += ks2; x1 += ks0 + 2u;
  TF_R(13) TF_R(15) TF_R(26) TF_R(6)   x0 += ks0; x1 += ks1 + 3u;
  TF_R(17) TF_R(29) TF_R(16) TF_R(24)  x0 += ks1; x1 += ks2 + 4u;
  TF_R(13) TF_R(15) TF_R(26) TF_R(6)   x0 += ks2; x1 += ks0 + 5u;
#undef TF_R
  o0 = x0; o1 = x1;
}

// ---------------------------------------------------------------------------
// Kernel 1: per (crystal, pick) selection. One 512-thread block per pick.
//   - count n via LDS atomic
//   - threefry-keyed bitonic sort of indices 0..511 => random permutation
//   - stable-compact values < n, first `cap`, pad with -1   -> sel[B][300]
// ---------------------------------------------------------------------------
__global__ __launch_bounds__(PATM) void select_kernel(
    const int* __restrict__ atom_num, const int* __restrict__ cai,
    int* __restrict__ sel) {
  const int b    = blockIdx.x >> 1;
  const int pick = blockIdx.x & 1;   // 0 = others (cap 180), 1 = carbon (cap 120)
  const int tid  = threadIdx.x;

  __shared__ int      s_n;
  __shared__ uint32_t skey[PATM];
  __shared__ int      sval[PATM];

  const int an   = atom_num[cai[b * PATM + tid]];
  const int flag = pick ? (an == 6) : ((an != 6) && (an != 1));

  if (tid == 0) s_n = 0;
  __syncthreads();
  if (flag) atomicAdd(&s_n, 1);

  // key chain: root(0,42) -> crystal key -> pick key -> per-index sort key
  uint32_t o0, o1, kb0, kb1, kp0, kp1;
  threefry2x32(0u, 42u, (uint32_t)(2 * b), (uint32_t)(2 * b + 1), kb0, kb1);
  threefry2x32(kb0, kb1, (uint32_t)(2 * pick), (uint32_t)(2 * pick + 1), kp0, kp1);
  threefry2x32(kp0, kp1, (uint32_t)tid, 0u, o0, o1);
  skey[tid] = o0;
  sval[tid] = tid;
  __syncthreads();

  const int n = s_n;

  // bitonic sort ascending by key (deterministic incl. ties)
  for (int k = 2; k <= PATM; k <<= 1) {
    for (int j = k >> 1; j > 0; j >>= 1) {
      const int ixj = tid ^ j;
      if (ixj > tid) {
        const uint32_t ka = skey[tid], kc = skey[ixj];
        const bool up = ((tid & k) == 0);
        if (up ? (ka > kc) : (ka < kc)) {
          skey[tid] = kc; skey[ixj] = ka;
          const int tv = sval[tid]; sval[tid] = sval[ixj]; sval[ixj] = tv;
        }
      }
      __syncthreads();
    }
  }

  if (tid == 0) {
    const int cap = pick ? CAP_C : CAP_O;
    const int off = b * SLOTS + (pick ? CAP_O : 0);
    int j = 0;
    for (int i = 0; i < PATM && j < cap; ++i) {
      const int v = sval[i];
      if (v < n) sel[off + j++] = v;      // permutation of {0..n-1}, truncated
    }
    for (; j < cap; ++j) sel[off + j] = -1;
  }
}

// ---------------------------------------------------------------------------
// Kernel 2: fused embedding gather for the 19200 selected rows.
// Per block: stage node_emb + edge_emb tables (61 KB) into LDS via CDNA5
// async global->LDS B128 copies (ASYNCcnt), then one wave per output row:
// 16 lanes x float4 cover one 64-float segment, 2 segments per 32 lanes.
// ---------------------------------------------------------------------------
__global__ __launch_bounds__(256) void gather_kernel(
    const int* __restrict__ atom_num, const int* __restrict__ nbr_idx,
    const float* __restrict__ nbr_fea, const int* __restrict__ cai,
    const float* __restrict__ node_emb, const float* __restrict__ edge_emb,
    const int* __restrict__ sel, float* __restrict__ out) {
  __shared__ float smem[SMEM_F];  // [0..7615] node table, [7616..15231] edge table

  // low 32 bits of a flat LDS pointer == workgroup-relative LDS byte address
  const uint32_t sbase = (uint32_t)(uintptr_t)(&smem[0]);
  const unsigned long long nsrc = (unsigned long long)(uintptr_t)node_emb;
  const unsigned long long esrc = (unsigned long long)(uintptr_t)edge_emb;

  for (int idx = threadIdx.x; idx < F4_TOT; idx += 256) {
    const uint32_t la = sbase + (uint32_t)idx * 16u;
    if (idx < F4_TAB) {
      const uint32_t go = (uint32_t)idx * 16u;
      asm volatile("global_load_async_to_lds_b128 %0, %1, %2"
                   :: "v"(la), "v"(go), "s"(nsrc) : "memory");
    } else {
      const uint32_t go = (uint32_t)(idx - F4_TAB) * 16u;
      asm volatile("global_load_async_to_lds_b128 %0, %1, %2"
                   :: "v"(la), "v"(go), "s"(esrc) : "memory");
    }
  }
  asm volatile("s_wait_asynccnt 0x0" ::: "memory");
  __syncthreads();

  const int lane = threadIdx.x & 31;
  const int wave = threadIdx.x >> 5;
  const int half = lane >> 4;        // which of the two segments this lane covers
  const int c4   = (lane & 15) * 4;  // float offset within a 64-float segment

  const int rbeg = blockIdx.x * 32;
  int rend = rbeg + 32; if (rend > ROWS) rend = ROWS;

  for (int r = rbeg + wave; r < rend; r += 8) {
    const int b  = r / SLOTS;
    const int sv = sel[r];
    float* orow = out + (size_t)r * HID;

    if (sv < 0) {
      float4 z; z.x = 0.f; z.y = 0.f; z.z = 0.f; z.w = 0.f;
      #pragma unroll
      for (int t = 0; t < 6; ++t) {
        const int m = 2 * t + half;
        *(float4*)(orow + m * DDIM + c4) = z;
      }
    } else {
      const int a   = cai[b * PATM + sv];
      const int ana = atom_num[a];
      const float4 nd = *(const float4*)(&smem[ana * DDIM + c4]);
      #pragma unroll
      for (int t = 0; t < 6; ++t) {
        const int m   = 2 * t + half;
        const int nb  = nbr_idx[a * MNBR + m];
        const int ane = atom_num[nb];
        const float4 f = *(const float4*)(nbr_fea + ((size_t)a * MNBR + m) * DDIM + c4);
        const float4 e = *(const float4*)(&smem[EMB_F + ane * DDIM + c4]);
        float4 o;
        o.x = f.x + e.x + nd.x;
        o.y = f.y + e.y + nd.y;
        o.z = f.z + e.z + nd.z;
        o.w = f.w + e.w + nd.w;
        *(float4*)(orow + m * DDIM + c4) = o;
      }
    }
  }
}

// ---------------------------------------------------------------------------
// Kernel 3: mask = (rowsum != 0). rowsum = G[19200x768] x ones[768] via
// V_WMMA_F32_16X16X4_F32. A-matrix ISA layout: lane m (0..15) holds rows,
// a.x -> K = half*2, a.y -> K = half*2+1  (half = lane>>4). B = all-ones so
// its layout is irrelevant. D[m][*] = rowsum(m); row m lives in VGPR m&7 of
// lane 0 (m<8) / lane 16 (m>=8).
//
// Out-of-range rows (300..303 of the last tile) are handled by CLAMPING the
// read row to 299 (unconditional, coalescible b64 loads; no EXEC toggling)
// and guarding only the mask writes. K-unroll x4 batches four b64 loads
// ahead of the four dependent WMMAs.
// ---------------------------------------------------------------------------
typedef __attribute__((ext_vector_type(2))) float v2f;
typedef __attribute__((ext_vector_type(8))) float v8f;

__global__ __launch_bounds__(32) void mask_kernel(const float* __restrict__ g,
                                                  float* __restrict__ mask) {
  const int b     = blockIdx.x / 19;
  const int t     = blockIdx.x % 19;
  const int rbase = t * 16;
  const int lane  = threadIdx.x;
  const int m     = lane & 15;
  const int half  = lane >> 4;
  int row = rbase + m;
  if (row > SLOTS - 1) row = SLOTS - 1;   // clamp: duplicate reads, writes guarded
  const float2* __restrict__ G2 =
      (const float2*)(g + ((size_t)b * SLOTS + row) * HID);

#if __has_builtin(__builtin_amdgcn_wmma_f32_16x16x4_f32)
  v8f c = {0.f, 0.f, 0.f, 0.f, 0.f, 0.f, 0.f, 0.f};
  v2f ones; ones.x = 1.f; ones.y = 1.f;
  #pragma unroll 2
  for (int k = 0; k < HID; k += 16) {           // 4 WMMA K-steps per iteration
    const int p = (k >> 1) + half;              // float2 index for this lane
    const float2 q0 = G2[p];
    const float2 q1 = G2[p + 2];
    const float2 q2 = G2[p + 4];
    const float2 q3 = G2[p + 6];
    v2f a0; a0.x = q0.x; a0.y = q0.y;
    v2f a1; a1.x = q1.x; a1.y = q1.y;
    v2f a2; a2.x = q2.x; a2.y = q2.y;
    v2f a3; a3.x = q3.x; a3.y = q3.y;
    c = __builtin_amdgcn_wmma_f32_16x16x4_f32(false, a0, false, ones, (short)0, c, false, false);
    c = __builtin_amdgcn_wmma_f32_16x16x4_f32(false, a1, false, ones, (short)0, c, false, false);
    c = __builtin_amdgcn_wmma_f32_16x16x4_f32(false, a2, false, ones, (short)0, c, false, false);
    c = __builtin_amdgcn_wmma_f32_16x16x4_f32(false, a3, false, ones, (short)0, c, false, false);
  }
  if ((lane & 15) == 0) {     // lane 0 -> rows rbase..+7, lane 16 -> rbase+8..+15
    #pragma unroll
    for (int v = 0; v < 8; ++v) {
      const int r2 = rbase + half * 8 + v;
      if (r2 < SLOTS) mask[b * SLOTS + r2] = (c[v] != 0.f) ? 1.f : 0.f;
    }
  }
#else
  if (half == 0 && rbase + m < SLOTS) {
    float s = 0.f;
    for (int k = 0; k < HID / 2; ++k) { s += G2[k].x + G2[k].y; }
    mask[b * SLOTS + rbase + m] = (s != 0.f) ? 1.f : 0.f;
  }
#endif
}

// ---------------------------------------------------------------------------
// Launcher
// ---------------------------------------------------------------------------
extern "C" void kernel_launch(void* const* d_in, const int* in_sizes, int n_in,
                              void* d_out, int out_size, void* d_ws, size_t ws_size,
                              hipStream_t stream) {
  (void)in_sizes; (void)n_in; (void)out_size; (void)ws_size;
  const int*   atom_num = (const int*)  d_in[0];
  const int*   nbr_idx  = (const int*)  d_in[1];
  const float* nbr_fea  = (const float*)d_in[2];
  const int*   cai      = (const int*)  d_in[3];
  const float* node_emb = (const float*)d_in[4];
  const float* edge_emb = (const float*)d_in[5];
  float* out  = (float*)d_out;                          // [64,300,768] then [64,300]
  float* mask = out + (size_t)ROWS * HID;
  int*   sel  = (int*)d_ws;                             // [64][300] selected rows

  select_kernel<<<BCRY * 2, PATM, 0, stream>>>(atom_num, cai, sel);
  gather_kernel<<<ROWS / 32, 256, 0, stream>>>(atom_num, nbr_idx, nbr_fea, cai,
                                               node_emb, edge_emb, sel, out);
  mask_kernel<<<BCRY * 19, 32, 0, stream>>>(out, mask);
}